// ExpertDispatcher_54546084659700
// MI455X (gfx1250) — compile-verified
//
#include <hip/hip_runtime.h>
#include <hip/hip_bf16.h>
#include <stdint.h>

// MoE expert dispatch for MI455X (gfx1250).
// B=4, S=4096, D=1024, K=2, E=16, capacity=2560.
// Pure data-movement op: ~64MB unique gather read (x2 reuse) + 168MB write.
// Strategy: deterministic order-preserving inverse map (expert,pos)->slot via
// per-expert prefix scan, then a single gather pass that writes every output
// element exactly once. Row copies use the CDNA5 async global->LDS engine
// (ASYNCcnt); output rows are stored non-temporally (TH=NT) so the 64MB input
// working set stays resident in the 192MB L2 for the ~2x gather reuse.

#define NE   16
#define CAP  2560
#define TK   32768      // T*K routed slots
#define DIM  1024
#define D4   (DIM / 4)  // 256 float4 per row

typedef float v4f __attribute__((ext_vector_type(4)));

// ---------------------------------------------------------------------------
// Kernel 1: build inv[e*CAP + p] = slot index (or -1 for empty/dropped).
// One block per expert, 256 threads; thread t owns 128 contiguous slots so the
// LDS prefix scan reproduces jnp.cumsum's flat-order position exactly.
// ---------------------------------------------------------------------------
__global__ __launch_bounds__(256) void moe_pos_kernel(
    const int* __restrict__ eidx, int* __restrict__ inv) {
  const int expert = blockIdx.x;   // 0..15
  const int t      = threadIdx.x;  // 0..255

  // Init this expert's slice of the inverse map (exclusively ours).
  for (int i = t; i < CAP; i += 256) inv[expert * CAP + i] = -1;

  // Phase 1: count matches in my 128 contiguous slots (int4 loads).
  const int   begin = t * (TK / 256);  // 128 slots per thread
  const int4* e4    = (const int4*)(eidx + begin);
  int cnt = 0;
#pragma unroll 8
  for (int j = 0; j < 32; ++j) {
    int4 v = e4[j];
    cnt += (v.x == expert) + (v.y == expert) + (v.z == expert) + (v.w == expert);
  }

  // Block-wide inclusive scan over 256 thread counts (flat order preserved).
  __shared__ int sc[256];
  sc[t] = cnt;
  __syncthreads();
  for (int off = 1; off < 256; off <<= 1) {
    int v = (t >= off) ? sc[t - off] : 0;
    __syncthreads();
    sc[t] += v;
    __syncthreads();
  }
  int p = sc[t] - cnt;  // exclusive base for my range

  // Phase 2: record ordered positions; overflow (>=CAP) dropped like the
  // reference's scratch column.
#pragma unroll 8
  for (int j = 0; j < 32; ++j) {
    int4 v = e4[j];
    int  s = begin + j * 4;
    if (v.x == expert) { if (p < CAP) inv[expert * CAP + p] = s;     ++p; }
    if (v.y == expert) { if (p < CAP) inv[expert * CAP + p] = s + 1; ++p; }
    if (v.z == expert) { if (p < CAP) inv[expert * CAP + p] = s + 2; ++p; }
    if (v.w == expert) { if (p < CAP) inv[expert * CAP + p] = s + 3; ++p; }
  }
}

// ---------------------------------------------------------------------------
// Kernel 2: one block per (expert, capacity-slot); writes every output element
// exactly once. Routed rows: async global->LDS stage (b128 per lane), wait on
// ASYNCcnt, then LDS->VGPR->non-temporal global store. Empty rows: direct
// non-temporal zero store. Branch is uniform per block (inv[b] is scalar), so
// EXEC is all-1s in the async path as the ISA requires.
// ---------------------------------------------------------------------------
__global__ __launch_bounds__(256) void moe_gather_kernel(
    const float* __restrict__ flat_in, const float* __restrict__ wts,
    const int* __restrict__ inv, float* __restrict__ exp_in,
    float* __restrict__ exp_w, float* __restrict__ exp_tok,
    float* __restrict__ exp_valid) {
  const int b   = blockIdx.x;   // 0 .. NE*CAP-1
  const int t   = threadIdx.x;  // 0 .. 255
  const int src = inv[b];       // routed slot or -1 (uniform per block)

  v4f* dst = (v4f*)(exp_in + (size_t)b * DIM);

  if (src >= 0) {
    const int token = src >> 1;  // K == 2
    if (t == 0) {
      exp_w[b]     = wts[src];
      exp_tok[b]   = (float)token;
      exp_valid[b] = 1.0f;
    }
    const v4f* srow = (const v4f*)(flat_in + (size_t)token * DIM);
#if defined(__gfx1250__)
    __shared__ v4f buf[D4];  // 4KB staging tile
    // Low 32 bits of the generic LDS address == workgroup-relative LDS byte
    // offset; hardware adds LDS_BASE (ISA 15.18 async addressing).
    unsigned ldsoff = (unsigned)(size_t)(&buf[t]);
    uint64_t gaddr  = (uint64_t)(size_t)(srow + t);
    asm volatile("global_load_async_to_lds_b128 %0, %1, off"
                 :: "v"(ldsoff), "v"(gaddr)
                 : "memory");
    asm volatile("s_wait_asynccnt 0" ::: "memory");
    v4f val = buf[t];  // ds_load_b128
    __builtin_nontemporal_store(val, &dst[t]);  // global_store_b128 TH=NT
#else
    __builtin_nontemporal_store(srow[t], &dst[t]);
#endif
  } else {
    if (t == 0) {
      exp_w[b]     = 0.0f;
      exp_tok[b]   = -1.0f;
      exp_valid[b] = 0.0f;
    }
    v4f z = {0.0f, 0.0f, 0.0f, 0.0f};
    __builtin_nontemporal_store(z, &dst[t]);
  }
}

// ---------------------------------------------------------------------------
extern "C" void kernel_launch(void* const* d_in, const int* in_sizes, int n_in,
                              void* d_out, int out_size, void* d_ws,
                              size_t ws_size, hipStream_t stream) {
  const float* inputs        = (const float*)d_in[0];  // [B,S,D] f32
  const int*   top_k_indices = (const int*)d_in[1];    // [B,S,K] i32
  const float* top_k_weights = (const float*)d_in[2];  // [B,S,K] f32

  // Outputs concatenated flat in reference return order (all float).
  float* exp_in    = (float*)d_out;                       // NE*CAP*DIM
  float* exp_w     = exp_in + (size_t)NE * CAP * DIM;     // NE*CAP
  float* exp_tok   = exp_w + (size_t)NE * CAP;            // NE*CAP
  float* exp_valid = exp_tok + (size_t)NE * CAP;          // NE*CAP

  int* inv = (int*)d_ws;  // NE*CAP ints = 160KB scratch, rebuilt every call

  moe_pos_kernel<<<NE, 256, 0, stream>>>(top_k_indices, inv);
  moe_gather_kernel<<<NE * CAP, 256, 0, stream>>>(
      inputs, top_k_weights, inv, exp_in, exp_w, exp_tok, exp_valid);
}